// PWLApproximation_42442866819367
// MI455X (gfx1250) — compile-verified
//
#include <hip/hip_runtime.h>

#define SEGMENTS 16

typedef float v4f __attribute__((ext_vector_type(4)));
typedef float v2f __attribute__((ext_vector_type(2)));

// Segment index exactly matching
//   idx = clip(searchsorted(linspace(-4,4,17), x, 'right') - 1, 0, 15)
// then slopes[idx]*x + intercepts[idx] via a conflict-free LDS float2 gather.
__device__ __forceinline__ float pwl_eval(float v, const v2f* __restrict__ lut) {
    float t = fmaf(v, 2.0f, 8.0f);          // (v + 4) * 2, single rounding
    t = fminf(fmaxf(t, -1.0f), 16.0f);      // keep int path overflow-safe
    int i = (int)floorf(t);
    float bp = fmaf((float)i, 0.5f, -4.0f); // exact breakpoint for bin i
    if (v < bp)              i -= 1;        // 1-ulp fixup -> searchsorted semantics
    else if (v >= bp + 0.5f) i += 1;
    i = i < 0 ? 0 : (i > SEGMENTS - 1 ? SEGMENTS - 1 : i);
    v2f si = lut[i];                        // ds_load_b64, conflict-free
    return fmaf(si.x, v, si.y);
}

__device__ __forceinline__ v4f pwl_eval4(v4f v, const v2f* __restrict__ lut) {
    v4f r;
    r.x = pwl_eval(v.x, lut);
    r.y = pwl_eval(v.y, lut);
    r.z = pwl_eval(v.z, lut);
    r.w = pwl_eval(v.w, lut);
    return r;
}

// Full-tile kernel: every block processes exactly 1024 float4s (4096 floats),
// no bounds checks, 32-bit offsets. Loads hoisted so they clause together.
__global__ __launch_bounds__(256) void pwl_main_kernel(
    const v4f* __restrict__ x4,
    const float* __restrict__ slopes,
    const float* __restrict__ intercepts,
    v4f* __restrict__ out4)
{
    __shared__ v2f lut[SEGMENTS];
    const unsigned int t = threadIdx.x;
    if (t < SEGMENTS) {
        v2f e;
        e.x = slopes[t];
        e.y = intercepts[t];
        lut[t] = e;
    }
    __syncthreads();

    const unsigned int base = blockIdx.x * 1024u + t;   // float4 index, fits 32-bit

    // Issue all four NT b128 loads back-to-back (one clause), then compute.
    v4f a = __builtin_nontemporal_load(&x4[base]);
    v4f b = __builtin_nontemporal_load(&x4[base + 256u]);
    v4f c = __builtin_nontemporal_load(&x4[base + 512u]);
    v4f d = __builtin_nontemporal_load(&x4[base + 768u]);

    v4f ra = pwl_eval4(a, lut);
    __builtin_nontemporal_store(ra, &out4[base]);
    v4f rb = pwl_eval4(b, lut);
    __builtin_nontemporal_store(rb, &out4[base + 256u]);
    v4f rc = pwl_eval4(c, lut);
    __builtin_nontemporal_store(rc, &out4[base + 512u]);
    v4f rd = pwl_eval4(d, lut);
    __builtin_nontemporal_store(rd, &out4[base + 768u]);
}

// Guarded scalar remainder kernel (covers [start, n)); tiny for the ref shape.
__global__ __launch_bounds__(256) void pwl_rem_kernel(
    const float* __restrict__ x,
    const float* __restrict__ slopes,
    const float* __restrict__ intercepts,
    float* __restrict__ out,
    long long start, long long n)
{
    __shared__ v2f lut[SEGMENTS];
    const unsigned int t = threadIdx.x;
    if (t < SEGMENTS) {
        v2f e;
        e.x = slopes[t];
        e.y = intercepts[t];
        lut[t] = e;
    }
    __syncthreads();

    long long i = start + (long long)blockIdx.x * blockDim.x + t;
    if (i < n) {
        out[i] = pwl_eval(x[i], lut);
    }
}

extern "C" void kernel_launch(void* const* d_in, const int* in_sizes, int n_in,
                              void* d_out, int out_size, void* d_ws, size_t ws_size,
                              hipStream_t stream) {
    const float* x          = (const float*)d_in[0];
    const float* slopes     = (const float*)d_in[1];
    const float* intercepts = (const float*)d_in[2];
    float* out              = (float*)d_out;

    const long long n  = (long long)in_sizes[0];
    const long long n4 = n >> 2;                 // whole float4 chunks

    const long long FULL = 1024;                 // float4s per full block
    const long long nFullBlocks = n4 / FULL;     // unguarded blocks

    if (nFullBlocks > 0) {
        pwl_main_kernel<<<(unsigned int)nFullBlocks, 256, 0, stream>>>(
            (const v4f*)x, slopes, intercepts, (v4f*)out);
    }

    const long long covered = nFullBlocks * FULL * 4;   // elements done
    if (covered < n) {
        const long long rem = n - covered;
        const unsigned int blocks = (unsigned int)((rem + 255) / 256);
        pwl_rem_kernel<<<blocks, 256, 0, stream>>>(
            x, slopes, intercepts, out, covered, n);
    }
}